// STANModel_62663572849068
// MI455X (gfx1250) — compile-verified
//
#include <hip/hip_runtime.h>
#include <hip/hip_bf16.h>

// ---------------------------------------------------------------------------
// STAN model: 16x { GAT(9->8x128) -> GAT(1024->128) -> mean-pool } -> GRU -> proj
// GEMM2 (h1[N,1024] @ W2[1024,128]) runs on V_WMMA_F32_16X16X4_F32 (wave32),
// with A/B tiles staged through LDS via GLOBAL_LOAD_ASYNC_TO_LDS_B128.
// ---------------------------------------------------------------------------

#define T_STEPS 16
#define F_INF   9
#define HIDD    128
#define HEADS1  8
#define D1      (HEADS1 * HIDD)   // 1024

typedef __attribute__((ext_vector_type(2))) float v2f;
typedef __attribute__((ext_vector_type(8))) float v8f;

// ---------------- utility ----------------
__global__ void k_zero(float* __restrict__ p, size_t n) {
  size_t i = (size_t)blockIdx.x * blockDim.x + threadIdx.x;
  if (i < n) p[i] = 0.0f;
}

// ---------------- GAT1 linear: x[N,9] @ W1[9,1024] -> h[N,1024] ----------------
__global__ void k_gemm1(const float* __restrict__ xt, const float* __restrict__ W1,
                        float* __restrict__ h, int N) {
  int idx = blockIdx.x * blockDim.x + threadIdx.x;
  if (idx >= N * D1) return;
  int n = idx / D1, j = idx % D1;
  const float* xr = xt + (size_t)n * F_INF;
  float acc = 0.f;
#pragma unroll
  for (int c = 0; c < F_INF; ++c) acc += xr[c] * W1[c * D1 + j];
  h[idx] = acc;
}

// ---------------- per-node attention logits: one wave per (node, head) ----------------
__global__ void k_attn_logits(const float* __restrict__ h,
                              const float* __restrict__ a_src,
                              const float* __restrict__ a_dst,
                              float* __restrict__ als, float* __restrict__ ald,
                              int N, int heads) {
  int wid  = (blockIdx.x * blockDim.x + threadIdx.x) >> 5;
  int lane = threadIdx.x & 31;
  if (wid >= N * heads) return;
  int hd = wid % heads;
  const float* row = h + (size_t)wid * HIDD;
  const float* as  = a_src + hd * HIDD;
  const float* ad  = a_dst + hd * HIDD;
  float s = 0.f, d = 0.f;
  for (int c = lane; c < HIDD; c += 32) { float v = row[c]; s += v * as[c]; d += v * ad[c]; }
#pragma unroll
  for (int off = 16; off; off >>= 1) { s += __shfl_down(s, off, 32); d += __shfl_down(d, off, 32); }
  if (lane == 0) { als[wid] = s; ald[wid] = d; }
}

// ---------------- edge pass 1: leaky-relu logit + ordered-uint atomic max ----------------
__global__ void k_edge_logit(const int* __restrict__ ei, int E, int N, int heads,
                             const float* __restrict__ als, const float* __restrict__ ald,
                             float* __restrict__ elog, unsigned int* __restrict__ segmax) {
  int idx = blockIdx.x * blockDim.x + threadIdx.x;
  int tot = (E + N) * heads;
  if (idx >= tot) return;
  int edge = idx / heads, hd = idx % heads;
  int s, d;
  if (edge < E) { s = ei[edge]; d = ei[E + edge]; } else { s = d = edge - E; }
  float v = als[(size_t)s * heads + hd] + ald[(size_t)d * heads + hd];
  v = (v > 0.f) ? v : 0.2f * v;                 // LeakyReLU(0.2)
  elog[idx] = v;
  unsigned int bits = __float_as_uint(v);
  unsigned int enc  = (bits & 0x80000000u) ? ~bits : (bits | 0x80000000u);
  atomicMax(&segmax[(size_t)d * heads + hd], enc);
}

// ---------------- edge pass 2: exp(logit - max[dst]) + segment sum ----------------
__global__ void k_edge_expsum(const int* __restrict__ ei, int E, int N, int heads,
                              const unsigned int* __restrict__ segmax,
                              float* __restrict__ elog, float* __restrict__ segsum) {
  int idx = blockIdx.x * blockDim.x + threadIdx.x;
  int tot = (E + N) * heads;
  if (idx >= tot) return;
  int edge = idx / heads, hd = idx % heads;
  int d = (edge < E) ? ei[E + edge] : (edge - E);
  unsigned int u = segmax[(size_t)d * heads + hd];
  float m = (u == 0u) ? 0.f
          : ((u & 0x80000000u) ? __uint_as_float(u & 0x7fffffffu) : __uint_as_float(~u));
  float ex = expf(elog[idx] - m);
  elog[idx] = ex;
  atomicAdd(&segsum[(size_t)d * heads + hd], ex);
}

// ---------------- edge pass 3: weighted scatter-sum, one wave per edge ----------------
__global__ void k_scatter(const int* __restrict__ ei, int E, int N, int heads, int dim,
                          const float* __restrict__ feat, const float* __restrict__ ealpha,
                          const float* __restrict__ segsum, float* __restrict__ agg) {
  int wid  = (blockIdx.x * blockDim.x + threadIdx.x) >> 5;
  int lane = threadIdx.x & 31;
  int tot  = E + N;
  if (wid >= tot) return;
  int s, d;
  if (wid < E) { s = ei[wid]; d = ei[E + wid]; } else { s = d = wid - E; }
  int rowlen = heads * dim;
  const float* fs = feat + (size_t)s * rowlen;
  float*       ag = agg  + (size_t)d * rowlen;
  for (int hd = 0; hd < heads; ++hd) {
    float alpha = ealpha[(size_t)wid * heads + hd] /
                  (segsum[(size_t)d * heads + hd] + 1e-16f);
    int base = hd * dim;
    for (int c = lane; c < dim; c += 32)
      atomicAdd(&ag[base + c], fs[base + c] * alpha);
  }
}

// ---------------- bias + relu (in place) ----------------
__global__ void k_bias_relu(float* __restrict__ p, const float* __restrict__ b,
                            int rowlen, size_t n) {
  size_t i = (size_t)blockIdx.x * blockDim.x + threadIdx.x;
  if (i >= n) return;
  float v = p[i] + b[i % rowlen];
  p[i] = v > 0.f ? v : 0.f;
}

// ---------------- GAT2 GEMM on WMMA: A[N,1024] @ B[1024,128] -> C[N,128] ------
// A tile (16 x 64) and B tile (64 x 128) staged in LDS with async copies.
#define KT 64
__global__ void __launch_bounds__(256) k_gemm2_wmma(const float* __restrict__ A,
                                                    const float* __restrict__ B,
                                                    float* __restrict__ C, int N) {
  __shared__ float As[16 * KT];       //  4 KB
  __shared__ float Bs[KT * HIDD];     // 32 KB
  const int tid  = threadIdx.x;
  const int wave = tid >> 5;                       // 0..7 -> 16-col tile
  const int lane = tid & 31;
  const int row0 = blockIdx.x * 16;
  const int col0 = wave * 16;
  const int m     = lane & 15;                     // A row / B col within tile
  const int khalf = (lane >> 4) * 2;               // K sub-pair select (0 or 2)
  v8f acc = {};

  // LDS byte offsets for the async-copy destination operand
  // (flat LDS aperture: addr[31:0] is the LDS address)
  const unsigned aBase = (unsigned)(size_t)(void*)As;
  const unsigned bBase = (unsigned)(size_t)(void*)Bs;

  // async A-tile source coords: 256 threads x 16B covers 16x64 floats
  const int ar  = tid >> 4;                        // 0..15 (tile row)
  const int ac4 = (tid & 15) * 4;                  // 0,4,..,60 (col quad)
  int arow = row0 + ar; if (arow >= N) arow = N - 1;   // clamp (N%16==0 in practice)

  for (int kc = 0; kc < D1; kc += KT) {
    __syncthreads();                               // prior chunk's LDS reads done
    // ---- async global -> LDS: A tile (1 x b128 per thread) ----
    {
      const float* gp = A + (size_t)arow * D1 + kc + ac4;
      unsigned la = aBase + (unsigned)((ar * KT + ac4) * 4);
      asm volatile("global_load_async_to_lds_b128 %0, %1, off"
                   :: "v"(la), "v"(gp) : "memory");
    }
    // ---- async global -> LDS: B tile (8 x b128 per thread) ----
#pragma unroll
    for (int it = 0; it < 8; ++it) {
      int idx = tid + it * 256;                    // 0..2047
      int br  = idx >> 5;                          // 0..63 (K row)
      int bc4 = (idx & 31) * 4;                    // 0,4,..,124
      const float* gp = B + (size_t)(kc + br) * HIDD + bc4;
      unsigned la = bBase + (unsigned)((br * HIDD + bc4) * 4);
      asm volatile("global_load_async_to_lds_b128 %0, %1, off"
                   :: "v"(la), "v"(gp) : "memory");
    }
    asm volatile("s_wait_asynccnt 0x0" ::: "memory");
    __syncthreads();                               // tiles visible to all waves

    // ---- 16 WMMA steps over this K chunk, all operands from LDS ----
#pragma unroll
    for (int kk = 0; kk < KT; kk += 4) {
      int k = kk + khalf;
      v2f a, b;
      a.x = As[m * KT + k];
      a.y = As[m * KT + k + 1];
      b.x = Bs[k * HIDD + col0 + m];
      b.y = Bs[(k + 1) * HIDD + col0 + m];
      acc = __builtin_amdgcn_wmma_f32_16x16x4_f32(
          /*neg_a=*/false, a, /*neg_b=*/false, b,
          /*c_mod=*/(short)0, acc, /*reuse_a=*/false, /*reuse_b=*/false);
    }
  }
  // C/D layout: lane<16 -> N=lane, M=vgpr ; lane>=16 -> N=lane-16, M=vgpr+8
  int ccol  = lane & 15;
  int rbase = (lane >> 4) * 8;
#pragma unroll
  for (int v = 0; v < 8; ++v) {
    int row = row0 + rbase + v;
    if (row < N) C[(size_t)row * HIDD + col0 + ccol] = acc[v];
  }
}

// ---------------- global mean pool over nodes ----------------
__global__ void k_meanpool(const float* __restrict__ h2, float* __restrict__ emb_t, int N) {
  int j = blockIdx.x;                              // feature index 0..127
  float s = 0.f;
  for (int n = threadIdx.x; n < N; n += blockDim.x) s += h2[(size_t)n * HIDD + j];
  __shared__ float red[256];
  red[threadIdx.x] = s; __syncthreads();
  for (int off = 128; off; off >>= 1) {
    if (threadIdx.x < off) red[threadIdx.x] += red[threadIdx.x + off];
    __syncthreads();
  }
  if (threadIdx.x == 0) emb_t[j] = red[0] / (float)N;
}

// ---------------- GRU (torch gate order r,z,n) + output proj; one block of 128 ----------------
__global__ void __launch_bounds__(HIDD) k_gru(const float* __restrict__ emb,
                      const float* __restrict__ w_ih, const float* __restrict__ w_hh,
                      const float* __restrict__ b_ih, const float* __restrict__ b_hh,
                      const float* __restrict__ w_out, const float* __restrict__ b_out,
                      float* __restrict__ out) {
  __shared__ float hsh[HIDD];
  __shared__ float red[HIDD];
  int j = threadIdx.x;
  hsh[j] = 0.f;
  __syncthreads();
  for (int t = 0; t < T_STEPS; ++t) {
    const float* xt = emb + t * HIDD;
    float gi[3], gh[3];
    for (int g = 0; g < 3; ++g) {
      int rowi = g * HIDD + j;
      float si = b_ih[rowi], sh = b_hh[rowi];
      for (int c = 0; c < HIDD; ++c) {
        si += w_ih[rowi * HIDD + c] * xt[c];
        sh += w_hh[rowi * HIDD + c] * hsh[c];
      }
      gi[g] = si; gh[g] = sh;
    }
    float r  = 1.f / (1.f + expf(-(gi[0] + gh[0])));
    float z  = 1.f / (1.f + expf(-(gi[1] + gh[1])));
    float nn = tanhf(gi[2] + r * gh[2]);
    float hnew = (1.f - z) * nn + z * hsh[j];
    __syncthreads();                               // all gate reads of hsh done
    hsh[j] = hnew;
    red[j] = hnew * w_out[j];
    __syncthreads();
    for (int off = 64; off; off >>= 1) {
      if (j < off) red[j] += red[j + off];
      __syncthreads();
    }
    if (j == 0) out[t] = red[0] + b_out[0];
    __syncthreads();
  }
}

// ---------------------------------------------------------------------------
extern "C" void kernel_launch(void* const* d_in, const int* in_sizes, int n_in,
                              void* d_out, int out_size, void* d_ws, size_t ws_size,
                              hipStream_t stream) {
  const float* x      = (const float*)d_in[0];
  const int*   ei     = (const int*)  d_in[1];
  const float* W1     = (const float*)d_in[2];
  const float* a_src1 = (const float*)d_in[3];
  const float* a_dst1 = (const float*)d_in[4];
  const float* b1     = (const float*)d_in[5];
  const float* W2     = (const float*)d_in[6];
  const float* a_src2 = (const float*)d_in[7];
  const float* a_dst2 = (const float*)d_in[8];
  const float* b2     = (const float*)d_in[9];
  const float* w_ih   = (const float*)d_in[10];
  const float* w_hh   = (const float*)d_in[11];
  const float* b_ih   = (const float*)d_in[12];
  const float* b_hh   = (const float*)d_in[13];
  const float* w_out  = (const float*)d_in[14];
  const float* b_out  = (const float*)d_in[15];
  float* out = (float*)d_out;

  const int N = in_sizes[0] / (T_STEPS * F_INF);   // 20000
  const int E = in_sizes[1] / 2;                   // 320000
  const int TOT = E + N;                           // edges + self loops

  // ---- workspace carve-up (floats) ----
  float* ws = (float*)d_ws;
  float* h_lin   = ws;                 ws += (size_t)N * D1;     // GAT1 linear out
  float* agg1    = ws;                 ws += (size_t)N * D1;     // GAT1 aggregate / h1
  float* g       = ws;                 ws += (size_t)N * HIDD;   // GAT2 linear out
  float* agg2    = ws;                 ws += (size_t)N * HIDD;   // GAT2 aggregate / h2
  float* als1    = ws;                 ws += (size_t)N * HEADS1;
  float* ald1    = ws;                 ws += (size_t)N * HEADS1;
  float* segsum1 = ws;                 ws += (size_t)N * HEADS1;
  unsigned int* segmax1 = (unsigned int*)ws; ws += (size_t)N * HEADS1;
  float* elog1   = ws;                 ws += (size_t)TOT * HEADS1;
  float* als2    = ws;                 ws += (size_t)N;
  float* ald2    = ws;                 ws += (size_t)N;
  float* segsum2 = ws;                 ws += (size_t)N;
  unsigned int* segmax2 = (unsigned int*)ws; ws += (size_t)N;
  float* elog2   = ws;                 ws += (size_t)TOT;
  float* emb     = ws;                 ws += (size_t)T_STEPS * HIDD;
  (void)ws_size; (void)n_in; (void)out_size;

  const int B = 256;
  auto blocks = [](size_t n, int b) { return (unsigned)((n + b - 1) / b); };

  const size_t nd1 = (size_t)N * D1;
  const size_t nh  = (size_t)N * HIDD;

  for (int t = 0; t < T_STEPS; ++t) {
    const float* xt = x + (size_t)t * N * F_INF;

    // ---- GAT layer 1 ----
    k_gemm1<<<blocks(nd1, B), B, 0, stream>>>(xt, W1, h_lin, N);
    k_attn_logits<<<blocks((size_t)N * HEADS1 * 32, B), B, 0, stream>>>(
        h_lin, a_src1, a_dst1, als1, ald1, N, HEADS1);
    k_zero<<<blocks((size_t)N * HEADS1, B), B, 0, stream>>>((float*)segmax1, (size_t)N * HEADS1);
    k_zero<<<blocks((size_t)N * HEADS1, B), B, 0, stream>>>(segsum1, (size_t)N * HEADS1);
    k_zero<<<blocks(nd1, B), B, 0, stream>>>(agg1, nd1);
    k_edge_logit<<<blocks((size_t)TOT * HEADS1, B), B, 0, stream>>>(
        ei, E, N, HEADS1, als1, ald1, elog1, segmax1);
    k_edge_expsum<<<blocks((size_t)TOT * HEADS1, B), B, 0, stream>>>(
        ei, E, N, HEADS1, segmax1, elog1, segsum1);
    k_scatter<<<blocks((size_t)TOT * 32, B), B, 0, stream>>>(
        ei, E, N, HEADS1, HIDD, h_lin, elog1, segsum1, agg1);
    k_bias_relu<<<blocks(nd1, B), B, 0, stream>>>(agg1, b1, D1, nd1);   // agg1 := h1

    // ---- GAT layer 2 (WMMA GEMM, async-LDS staged) ----
    k_gemm2_wmma<<<(N + 15) / 16, 256, 0, stream>>>(agg1, W2, g, N);
    k_attn_logits<<<blocks((size_t)N * 32, B), B, 0, stream>>>(
        g, a_src2, a_dst2, als2, ald2, N, 1);
    k_zero<<<blocks((size_t)N, B), B, 0, stream>>>((float*)segmax2, (size_t)N);
    k_zero<<<blocks((size_t)N, B), B, 0, stream>>>(segsum2, (size_t)N);
    k_zero<<<blocks(nh, B), B, 0, stream>>>(agg2, nh);
    k_edge_logit<<<blocks((size_t)TOT, B), B, 0, stream>>>(
        ei, E, N, 1, als2, ald2, elog2, segmax2);
    k_edge_expsum<<<blocks((size_t)TOT, B), B, 0, stream>>>(
        ei, E, N, 1, segmax2, elog2, segsum2);
    k_scatter<<<blocks((size_t)TOT * 32, B), B, 0, stream>>>(
        ei, E, N, 1, HIDD, g, elog2, segsum2, agg2);
    k_bias_relu<<<blocks(nh, B), B, 0, stream>>>(agg2, b2, HIDD, nh);   // agg2 := h2

    // ---- global mean pool -> emb[t] ----
    k_meanpool<<<HIDD, 256, 0, stream>>>(agg2, emb + (size_t)t * HIDD, N);
  }

  // ---- GRU + output projection ----
  k_gru<<<1, HIDD, 0, stream>>>(emb, w_ih, w_hh, b_ih, b_hh, w_out, b_out, out);
}